// GNNEncoder_74612171866326
// MI455X (gfx1250) — compile-verified
//
#include <hip/hip_runtime.h>

#define NN 50000
#define EE 800000
#define ET (EE + NN)        // edges + self loops
#define HH 3
#define DD 64
#define NEG_SLOPE 0.2f

typedef float v2f __attribute__((ext_vector_type(2)));
typedef float v8f __attribute__((ext_vector_type(8)));

// ---- ordered-uint encoding of float for atomicMax-based segment max ----
__device__ __forceinline__ unsigned f2ord(float f) {
  unsigned u = __float_as_uint(f);
  return (u & 0x80000000u) ? ~u : (u | 0x80000000u);
}
__device__ __forceinline__ float ord2f(unsigned u) {
  unsigned v = (u & 0x80000000u) ? (u & 0x7FFFFFFFu) : ~u;
  return __uint_as_float(v);
}

// ---- layer-1 projection (Din = 2, trivial) ----
__global__ __launch_bounds__(256) void gat_proj1(const float* __restrict__ x,
                                                 const float* __restrict__ W,
                                                 float* __restrict__ h) {
  int t = blockIdx.x * blockDim.x + threadIdx.x;       // over N*H*D
  if (t >= NN * HH * DD) return;
  int f  = t % DD;
  int r  = t / DD;
  int hd = r % HH;
  int n  = r / HH;
  h[t] = x[n * 2 + 0] * W[(hd * 2 + 0) * DD + f] +
         x[n * 2 + 1] * W[(hd * 2 + 1) * DD + f];
}

// ---- layers 2-4 projection: h[n,hd,f] = sum_k x[n,k] * W[hd,k,f] via WMMA fp32 ----
// one wave per (16-row node tile, head, 16-col feature tile); K=64 in steps of 4.
#define MTILES (NN / 16)          // 3125 (exact)
#define NTILES (DD / 16)          // 4
#define TOT_WAVES (MTILES * HH * NTILES)   // 37500
__global__ __launch_bounds__(256) void gat_proj_wmma(const float* __restrict__ x,
                                                     const float* __restrict__ W,
                                                     float* __restrict__ h) {
  int wave = (blockIdx.x * blockDim.x + threadIdx.x) >> 5;
  if (wave >= TOT_WAVES) return;                 // wave-uniform: EXEC stays all-1s
  int lane  = threadIdx.x & 31;
  int mt    = wave % MTILES;
  int rest  = wave / MTILES;
  int nt    = rest & (NTILES - 1);
  int head  = rest >> 2;

  int lhalf = lane & 15;
  int hi2   = (lane >> 4) * 2;                   // lanes 16-31 carry K+2, K+3
  const float* xr = x + (size_t)(mt * 16 + lhalf) * DD;
  const float* Wh = W + (size_t)head * DD * DD;
  int ncol = nt * 16 + lhalf;

  v8f acc = {};
  #pragma unroll
  for (int k = 0; k < DD; k += 4) {
    v2f a, b;
    a[0] = xr[k + hi2];                          // A: 16x4, M=lhalf, K=k+hi2(+1)
    a[1] = xr[k + hi2 + 1];
    b[0] = Wh[(k + hi2) * DD + ncol];            // B: 4x16, K rows, N=ncol
    b[1] = Wh[(k + hi2 + 1) * DD + ncol];
    acc = __builtin_amdgcn_wmma_f32_16x16x4_f32(
        /*neg_a=*/false, a, /*neg_b=*/false, b,
        /*c_mod=*/(short)0, acc, /*reuse_a=*/false, /*reuse_b=*/false);
  }
  // D layout: VGPR r -> row (r + 8*hi), col = lhalf
  int mbase = mt * 16 + (lane >> 4) * 8;
  #pragma unroll
  for (int r = 0; r < 8; ++r)
    h[((size_t)(mbase + r) * HH + head) * DD + ncol] = acc[r];
}

// ---- per-(node,head): attention logits; init segment-max & denom ----
__global__ __launch_bounds__(256) void gat_node_prep(const float* __restrict__ h,
                                                     const float* __restrict__ asrc,
                                                     const float* __restrict__ adst,
                                                     float* __restrict__ als,
                                                     float* __restrict__ ald,
                                                     unsigned* __restrict__ mu,
                                                     float* __restrict__ denom) {
  int t = blockIdx.x * blockDim.x + threadIdx.x;   // over N*H
  if (t >= NN * HH) return;
  int hd = t % HH, n = t / HH;
  const float* hp = h + ((size_t)n * HH + hd) * DD;
  const float* as = asrc + hd * DD;
  const float* ad = adst + hd * DD;
  float s = 0.f, d = 0.f;
  #pragma unroll 8
  for (int f = 0; f < DD; ++f) { float v = hp[f]; s += v * as[f]; d += v * ad[f]; }
  als[t] = s; ald[t] = d;
  mu[t] = 0u;            // below ord-encoding of any finite float we produce
  denom[t] = 0.f;
}

__global__ __launch_bounds__(256) void gat_zero(float* __restrict__ p, int n) {
  int t = blockIdx.x * blockDim.x + threadIdx.x;
  if (t < n) p[t] = 0.f;
}

// ---- per-edge: e = leaky_relu(al_s[src]+al_d[dst]); segment max via atomicMax ----
__global__ __launch_bounds__(256) void gat_edge_max(const int* __restrict__ ei,
                                                    const float* __restrict__ als,
                                                    const float* __restrict__ ald,
                                                    float* __restrict__ ebuf,
                                                    unsigned* __restrict__ mu) {
  int et = blockIdx.x * blockDim.x + threadIdx.x;
  if (et >= ET) return;
  int s, d;
  if (et < EE) { s = ei[et]; d = ei[EE + et]; } else { s = d = et - EE; }
  #pragma unroll
  for (int hd = 0; hd < HH; ++hd) {
    float e = als[s * HH + hd] + ald[d * HH + hd];
    e = (e >= 0.f) ? e : NEG_SLOPE * e;
    ebuf[(size_t)et * HH + hd] = e;
    atomicMax(&mu[d * HH + hd], f2ord(e));
  }
}

// ---- per-edge: w = exp(e - m[dst]); denom[dst] += w ----
__global__ __launch_bounds__(256) void gat_edge_sum(const int* __restrict__ ei,
                                                    const unsigned* __restrict__ mu,
                                                    float* __restrict__ ebuf,
                                                    float* __restrict__ denom) {
  int et = blockIdx.x * blockDim.x + threadIdx.x;
  if (et >= ET) return;
  int d = (et < EE) ? ei[EE + et] : (et - EE);
  #pragma unroll
  for (int hd = 0; hd < HH; ++hd) {
    float w = expf(ebuf[(size_t)et * HH + hd] - ord2f(mu[d * HH + hd]));
    ebuf[(size_t)et * HH + hd] = w;
    atomicAdd(&denom[d * HH + hd], w);
  }
}

// ---- per-edge: alpha = w / (denom[dst] + 1e-16) ----
__global__ __launch_bounds__(256) void gat_edge_alpha(const int* __restrict__ ei,
                                                      const float* __restrict__ denom,
                                                      float* __restrict__ ebuf) {
  int et = blockIdx.x * blockDim.x + threadIdx.x;
  if (et >= ET) return;
  int d = (et < EE) ? ei[EE + et] : (et - EE);
  #pragma unroll
  for (int hd = 0; hd < HH; ++hd)
    ebuf[(size_t)et * HH + hd] /= (denom[d * HH + hd] + 1e-16f);
}

// ---- 64 lanes per edge: acc[dst,hd,f] += alpha * h[src,hd,f] (coalesced, L2 atomics) ----
__global__ __launch_bounds__(256) void gat_edge_scatter(const int* __restrict__ ei,
                                                        const float* __restrict__ h,
                                                        const float* __restrict__ ebuf,
                                                        float* __restrict__ acc) {
  int t = blockIdx.x * blockDim.x + threadIdx.x;   // over ET*DD (= 54.4M)
  if (t >= ET * DD) return;
  int f  = t & (DD - 1);
  int et = t >> 6;
  int s, d;
  if (et < EE) { s = ei[et]; d = ei[EE + et]; } else { s = d = et - EE; }
  #pragma unroll
  for (int hd = 0; hd < HH; ++hd) {
    float alpha = ebuf[(size_t)et * HH + hd];
    float v = h[((size_t)s * HH + hd) * DD + f] * alpha;
    atomicAdd(&acc[((size_t)d * HH + hd) * DD + f], v);
  }
}

// ---- head-mean + bias + relu ----
__global__ __launch_bounds__(256) void gat_finalize(const float* __restrict__ acc,
                                                    const float* __restrict__ b,
                                                    float* __restrict__ out) {
  int t = blockIdx.x * blockDim.x + threadIdx.x;   // over N*DD
  if (t >= NN * DD) return;
  int f = t & (DD - 1);
  int n = t >> 6;
  float v = (acc[((size_t)n * HH + 0) * DD + f] +
             acc[((size_t)n * HH + 1) * DD + f] +
             acc[((size_t)n * HH + 2) * DD + f]) * (1.f / 3.f) + b[f];
  out[t] = v > 0.f ? v : 0.f;
}

extern "C" void kernel_launch(void* const* d_in, const int* in_sizes, int n_in,
                              void* d_out, int out_size, void* d_ws, size_t ws_size,
                              hipStream_t stream) {
  const float* x0 = (const float*)d_in[0];
  const int*   ei = (const int*)d_in[1];
  const float* Wt[4] = {(const float*)d_in[2],  (const float*)d_in[6],
                        (const float*)d_in[10], (const float*)d_in[14]};
  const float* As[4] = {(const float*)d_in[3],  (const float*)d_in[7],
                        (const float*)d_in[11], (const float*)d_in[15]};
  const float* Ad[4] = {(const float*)d_in[4],  (const float*)d_in[8],
                        (const float*)d_in[12], (const float*)d_in[16]};
  const float* Bs[4] = {(const float*)d_in[5],  (const float*)d_in[9],
                        (const float*)d_in[13], (const float*)d_in[17]};

  // workspace carve-out (~102 MB)
  float*    xbuf  = (float*)d_ws;                          // N*D
  float*    hbuf  = xbuf + (size_t)NN * DD;                // N*H*D
  float*    accb  = hbuf + (size_t)NN * HH * DD;           // N*H*D
  float*    als   = accb + (size_t)NN * HH * DD;           // N*H
  float*    ald   = als + (size_t)NN * HH;                 // N*H
  unsigned* mu    = (unsigned*)(ald + (size_t)NN * HH);    // N*H
  float*    denom = (float*)(mu + (size_t)NN * HH);        // N*H
  float*    ebuf  = denom + (size_t)NN * HH;               // ET*H

  const int B = 256;
  for (int L = 0; L < 4; ++L) {
    if (L == 0)
      gat_proj1<<<(NN * HH * DD + B - 1) / B, B, 0, stream>>>(x0, Wt[0], hbuf);
    else
      gat_proj_wmma<<<(TOT_WAVES * 32 + B - 1) / B, B, 0, stream>>>(xbuf, Wt[L], hbuf);

    gat_node_prep<<<(NN * HH + B - 1) / B, B, 0, stream>>>(hbuf, As[L], Ad[L],
                                                           als, ald, mu, denom);
    gat_zero<<<(NN * HH * DD + B - 1) / B, B, 0, stream>>>(accb, NN * HH * DD);
    gat_edge_max<<<(ET + B - 1) / B, B, 0, stream>>>(ei, als, ald, ebuf, mu);
    gat_edge_sum<<<(ET + B - 1) / B, B, 0, stream>>>(ei, mu, ebuf, denom);
    gat_edge_alpha<<<(ET + B - 1) / B, B, 0, stream>>>(ei, denom, ebuf);
    gat_edge_scatter<<<(ET * DD + B - 1) / B, B, 0, stream>>>(ei, hbuf, ebuf, accb);
    gat_finalize<<<(NN * DD + B - 1) / B, B, 0, stream>>>(accb, Bs[L],
        (L == 3) ? (float*)d_out : xbuf);
  }
}